// Model_23562190586223
// MI455X (gfx1250) — compile-verified
//
#include <hip/hip_runtime.h>
#include <hip/hip_bf16.h>

// Problem constants (from reference): B=8, H=W=2048, single channel, fp32.
#define BATCH 8
#define HDIM  2048
#define WDIM  2048

// Fused-stencil tiling: 64x64 outputs per workgroup, 256 threads (8 wave32).
// x/mask staging tile: rows [oy-2, oy+66), cols [ox-4, ox+68)  -> 68 x 72
// h tile: rows [oy-1, oy+65), cols [ox-1, ox+65) -> 66 rows, stride 68 (16B rows)
#define TILE 64
#define SXH  68            // x tile rows (TILE + 4)
#define SXW  72            // x tile cols (TILE + 8); row stride 288B (16B mult)
#define SXC  (SXW / 4)     // 16B chunks per x-tile row (18)
#define SHR  66            // h tile rows
#define SHW  68            // h tile row stride (= 17 x 16B, keeps float4 aligned)
#define NTHREADS 256

#define GLOBAL_AS __attribute__((address_space(1)))
#define LOCAL_AS  __attribute__((address_space(3)))

typedef int   v4i   __attribute__((ext_vector_type(4)));
typedef int   v2i   __attribute__((ext_vector_type(2)));
typedef float f32x4 __attribute__((ext_vector_type(4)));
typedef float f32x2 __attribute__((ext_vector_type(2)));

// ---- CDNA5 async global->LDS copies (ASYNCcnt-tracked) ----------------------
__device__ __forceinline__ void async_copy_b32(const void* gptr, void* lptr) {
#if __has_builtin(__builtin_amdgcn_global_load_async_to_lds_b32)
    __builtin_amdgcn_global_load_async_to_lds_b32(
        (GLOBAL_AS int*)gptr, (LOCAL_AS int*)lptr, /*offset=*/0, /*cpol=*/0);
#else
    unsigned lds_addr = (unsigned)(size_t)(LOCAL_AS char*)lptr;
    unsigned long long ga = (unsigned long long)gptr;
    asm volatile("global_load_async_to_lds_b32 %0, %1, off"
                 :: "v"(lds_addr), "v"(ga)
                 : "memory");
#endif
}

// 16-byte async copy; caller guarantees 16B alignment of both addresses.
__device__ __forceinline__ void async_copy_b128(const void* gptr, void* lptr) {
#if __has_builtin(__builtin_amdgcn_global_load_async_to_lds_b128)
    __builtin_amdgcn_global_load_async_to_lds_b128(
        (GLOBAL_AS v4i*)gptr, (LOCAL_AS v4i*)lptr, /*offset=*/0, /*cpol=*/0);
#else
    const int* g = (const int*)gptr;
    int*       l = (int*)lptr;
    async_copy_b32(g + 0, l + 0);
    async_copy_b32(g + 1, l + 1);
    async_copy_b32(g + 2, l + 2);
    async_copy_b32(g + 3, l + 3);
#endif
}

__device__ __forceinline__ void wait_async_zero() {
#if __has_builtin(__builtin_amdgcn_s_wait_asynccnt)
    __builtin_amdgcn_s_wait_asynccnt(0);
#else
    asm volatile("s_wait_asynccnt 0x0" ::: "memory");
#endif
}

// ---- Fused masked conv3x3 -> relu -> conv3x3 --------------------------------
__global__ __launch_bounds__(NTHREADS)
void mink_fused_kernel(const float* __restrict__ x,
                       const int*   __restrict__ mask,
                       const float* __restrict__ w1,
                       const float* __restrict__ b1,
                       const float* __restrict__ w2,
                       const float* __restrict__ b2,
                       float* __restrict__ out) {
    __shared__ __align__(16) float s_x[SXH * SXW];  // x tile, becomes x*m in place
    __shared__ __align__(16) int   s_m[SXH * SXW];  // occupancy tile (int throughout)
    __shared__ __align__(16) float s_h[SHR * SHW];  // intermediate h tile

    const int tid = threadIdx.x;
    const int ox  = blockIdx.x * TILE;
    const int oy  = blockIdx.y * TILE;
    const long long base = (long long)blockIdx.z * (long long)HDIM * (long long)WDIM;

    // Blocks whose staging tile can poke outside the image (block-uniform).
    const bool edge = (ox == 0) | (oy == 0) | (ox + TILE == WDIM) | (oy + TILE == HDIM);

    if (edge) {
        // Zero the staging tiles so the out-of-image halo contributes nothing.
        const f32x4 zf = {0.f, 0.f, 0.f, 0.f};
        const v4i   zi = {0, 0, 0, 0};
        for (int i = tid; i < (SXH * SXW) / 4; i += NTHREADS) {
            *(f32x4*)&s_x[4 * i] = zf;
            *(v4i*)  &s_m[4 * i] = zi;
        }
        __syncthreads();  // drain DS stores before async writes can land
    }

    // Stage 1: async DMA the halo tile of x and mask into LDS, 16B per request.
    for (int i = tid; i < SXH * SXC; i += NTHREADS) {
        const int row = i / SXC;
        const int k   = i - row * SXC;
        const int gy  = oy - 2 + row;
        const int gc  = ox - 4 + 4 * k;   // 16B-aligned global column
        if (!edge || ((unsigned)gy < (unsigned)HDIM && (unsigned)gc < (unsigned)WDIM)) {
            const long long g = base + (long long)gy * WDIM + gc;
            const int l = row * SXW + 4 * k;
            async_copy_b128(x + g,    &s_x[l]);
            async_copy_b128(mask + g, &s_m[l]);
        }
    }

    // Uniform weights: scalar loads overlap the async copies.
    float W1[9], W2[9];
#pragma unroll
    for (int kk = 0; kk < 9; ++kk) { W1[kk] = w1[kk]; W2[kk] = w2[kk]; }
    const float B1 = b1[0];
    const float B2 = b2[0];

    wait_async_zero();   // this wave's async LDS writes complete
    __syncthreads();     // all waves' writes visible

    // Stage 2: apply occupancy mask in place: s_x = x * m (float4-wide).
    for (int i = tid; i < (SXH * SXW) / 4; i += NTHREADS) {
        const v4i   mi = *(const v4i*)  &s_m[4 * i];
        f32x4       xv = *(const f32x4*)&s_x[4 * i];
        xv.x *= (float)mi.x;
        xv.y *= (float)mi.y;
        xv.z *= (float)mi.z;
        xv.w *= (float)mi.w;
        *(f32x4*)&s_x[4 * i] = xv;
    }
    __syncthreads();

    // Stage 3: h = relu((conv3x3(x*m) + b1) * m); 4 h-points per thread-chunk.
    // 66 rows x 17 chunks; the last chunk's top 2 lanes land in h padding.
    for (int i = tid; i < SHR * 17; i += NTHREADS) {
        const int r  = i / 17;
        const int cb = (i - r * 17) * 4;          // h column base (0..64)
        const float* p = &s_x[r * SXW + cb + 2];  // 8B-aligned
        float v[3][6];
#pragma unroll
        for (int rr = 0; rr < 3; ++rr) {
            const f32x2 a = *(const f32x2*)(p + rr * SXW);
            const f32x2 b = *(const f32x2*)(p + rr * SXW + 2);
            const f32x2 c = *(const f32x2*)(p + rr * SXW + 4);
            v[rr][0] = a.x; v[rr][1] = a.y; v[rr][2] = b.x;
            v[rr][3] = b.y; v[rr][4] = c.x; v[rr][5] = c.y;
        }
        const int* mrow = &s_m[(r + 1) * SXW + cb + 2];  // centers at cb+3..cb+6
        const v2i m0 = *(const v2i*)(mrow);
        const v2i m1 = *(const v2i*)(mrow + 2);
        const v2i m2 = *(const v2i*)(mrow + 4);
        const float mf[4] = {(float)m0.y, (float)m1.x, (float)m1.y, (float)m2.x};
        f32x4 hv;
#pragma unroll
        for (int j = 0; j < 4; ++j) {
            float acc = W1[0] * v[0][j] + W1[1] * v[0][j + 1] + W1[2] * v[0][j + 2]
                      + W1[3] * v[1][j] + W1[4] * v[1][j + 1] + W1[5] * v[1][j + 2]
                      + W1[6] * v[2][j] + W1[7] * v[2][j + 1] + W1[8] * v[2][j + 2];
            const float hval = fmaxf((acc + B1) * mf[j], 0.0f);
            if (j == 0) hv.x = hval;
            if (j == 1) hv.y = hval;
            if (j == 2) hv.z = hval;
            if (j == 3) hv.w = hval;
        }
        *(f32x4*)&s_h[r * SHW + cb] = hv;  // 16B-aligned
    }
    __syncthreads();

    // Stage 4: out = (conv3x3(h) + b2) * m; 4 outputs per thread-chunk,
    // 128-bit global stores (ox and cb are multiples of 4).
    for (int i = tid; i < TILE * (TILE / 4); i += NTHREADS) {
        const int r  = i >> 4;
        const int cb = (i & 15) * 4;
        const float* q = &s_h[r * SHW + cb];
        float v[3][6];
#pragma unroll
        for (int rr = 0; rr < 3; ++rr) {
            const f32x4 a = *(const f32x4*)(q + rr * SHW);      // 16B-aligned
            const f32x2 b = *(const f32x2*)(q + rr * SHW + 4);  // 8B-aligned
            v[rr][0] = a.x; v[rr][1] = a.y; v[rr][2] = a.z;
            v[rr][3] = a.w; v[rr][4] = b.x; v[rr][5] = b.y;
        }
        const v4i mi = *(const v4i*)&s_m[(r + 2) * SXW + cb + 4];  // 16B-aligned
        const float mf[4] = {(float)mi.x, (float)mi.y, (float)mi.z, (float)mi.w};
        f32x4 ov;
#pragma unroll
        for (int j = 0; j < 4; ++j) {
            float acc = W2[0] * v[0][j] + W2[1] * v[0][j + 1] + W2[2] * v[0][j + 2]
                      + W2[3] * v[1][j] + W2[4] * v[1][j + 1] + W2[5] * v[1][j + 2]
                      + W2[6] * v[2][j] + W2[7] * v[2][j + 1] + W2[8] * v[2][j + 2];
            const float oval = (acc + B2) * mf[j];
            if (j == 0) ov.x = oval;
            if (j == 1) ov.y = oval;
            if (j == 2) ov.z = oval;
            if (j == 3) ov.w = oval;
        }
        *(f32x4*)&out[base + (long long)(oy + r) * WDIM + (ox + cb)] = ov;
    }
}

extern "C" void kernel_launch(void* const* d_in, const int* in_sizes, int n_in,
                              void* d_out, int out_size, void* d_ws, size_t ws_size,
                              hipStream_t stream) {
    const float* x    = (const float*)d_in[0];
    const int*   mask = (const int*)d_in[1];
    const float* w1   = (const float*)d_in[2];
    const float* b1   = (const float*)d_in[3];
    const float* w2   = (const float*)d_in[4];
    const float* b2   = (const float*)d_in[5];
    float* out = (float*)d_out;

    dim3 grid(WDIM / TILE, HDIM / TILE, BATCH);
    mink_fused_kernel<<<grid, NTHREADS, 0, stream>>>(x, mask, w1, b1, w2, b2, out);
}